// NNLQP_4922032521409
// MI455X (gfx1250) — compile-verified
//
#include <hip/hip_runtime.h>
#include <hip/hip_bf16.h>

#define N_NODES   100000
#define N_EDGES   1600000
#define N_GRAPHS  128
#define IN_FEAT   44
#define GNN_H     128
#define FC_H      512
#define CAT_W     (GNN_H + 4)   // 132

typedef float v2f __attribute__((ext_vector_type(2)));
typedef float v8f __attribute__((ext_vector_type(8)));

// ---------------------------------------------------------------------------
// FP32 WMMA: D = A(16x4) * B(4x16) + C(16x16), full fp32 path on gfx1250.
// ---------------------------------------------------------------------------
__device__ inline v8f wmma4(float ax, float ay, float bx, float by, v8f c) {
    v2f a, b;
    a.x = ax; a.y = ay;
    b.x = bx; b.y = by;
    return __builtin_amdgcn_wmma_f32_16x16x4_f32(false, a, false, b,
                                                 (short)0, c, false, false);
}

// ---------------------------------------------------------------------------
// Utility kernels
// ---------------------------------------------------------------------------
__global__ void zero_f32(float* __restrict__ p, long long n) {
    long long i = (long long)blockIdx.x * blockDim.x + threadIdx.x;
    long long stride = (long long)gridDim.x * blockDim.x;
    for (; i < n; i += stride) p[i] = 0.0f;
}

__global__ void deg_count(const int* __restrict__ dst, float* __restrict__ deg) {
    int e = blockIdx.x * blockDim.x + threadIdx.x;
    if (e < N_EDGES) atomicAdd(&deg[dst[e]], 1.0f);
}

__global__ void invdeg_kernel(float* __restrict__ deg) {
    int i = blockIdx.x * blockDim.x + threadIdx.x;
    if (i < N_NODES) deg[i] = 1.0f / fmaxf(deg[i], 1.0f);
}

__global__ void copy_static(const float* __restrict__ sf, float* __restrict__ gnn_cat) {
    int t = blockIdx.x * blockDim.x + threadIdx.x;
    if (t < N_GRAPHS * 4) {
        int g = t >> 2, j = t & 3;
        gnn_cat[g * CAT_W + GNN_H + j] = sf[t];
    }
}

// Scatter-add of source features to destination accumulator (mean-agg numerator).
// Consecutive threads cover consecutive features of one edge -> coalesced
// gathers (512B per edge, layer 2) and coalesced atomic bursts. This is the
// HBM/atomic-bound hot spot (~1.6 GB of irreducible random traffic).
template <int F>
__global__ void scatter_add(const float* __restrict__ h, const int* __restrict__ src,
                            const int* __restrict__ dst, float* __restrict__ agg) {
    long long tid = (long long)blockIdx.x * blockDim.x + threadIdx.x;
    const long long total = (long long)N_EDGES * F;
    if (tid >= total) return;
    int e = (int)(tid / F);
    int f = (int)(tid - (long long)e * F);
    atomicAdd(&agg[(long long)dst[e] * F + f], h[(long long)src[e] * F + f]);
}

// ---------------------------------------------------------------------------
// SAGE layer GEMM with LDS-staged A tiles.
// Block = one M-tile (16 rows); 8 waves span N=128 (one 16x16 tile each).
// The 16xK self-features and invdeg-scaled agg-features are loaded once,
// coalesced, into LDS (row stride K+2 to dodge 16-way bank conflicts), then
// each wave feeds WMMA fragments from LDS (ds_load_b64 per lane).
// ---------------------------------------------------------------------------
template <int K>
__device__ inline v8f sage_tile_lds(const float* __restrict__ lds1,
                                    const float* __restrict__ lds2,
                                    const float* __restrict__ Ws,
                                    const float* __restrict__ Wn,
                                    int n0) {
    constexpr int LDK = K + 2;
    const int lane = threadIdx.x & 31;
    const int half = lane >> 4;
    const int l16  = lane & 15;
    const int bc   = n0 + l16;
    v8f acc = {0.f, 0.f, 0.f, 0.f, 0.f, 0.f, 0.f, 0.f};
    #pragma unroll
    for (int k0 = 0; k0 < K; k0 += 4) {
        const int ka = k0 + 2 * half;
        acc = wmma4(lds1[l16 * LDK + ka], lds1[l16 * LDK + ka + 1],
                    Ws[ka * GNN_H + bc], Ws[(ka + 1) * GNN_H + bc], acc);
    }
    #pragma unroll
    for (int k0 = 0; k0 < K; k0 += 4) {
        const int ka = k0 + 2 * half;
        acc = wmma4(lds2[l16 * LDK + ka], lds2[l16 * LDK + ka + 1],
                    Wn[ka * GNN_H + bc], Wn[(ka + 1) * GNN_H + bc], acc);
    }
    return acc;
}

template <int K>
__device__ inline void sage_stage(float* __restrict__ lds1, float* __restrict__ lds2,
                                  const float* __restrict__ A1,
                                  const float* __restrict__ A2,
                                  const float* __restrict__ invdeg, int m0) {
    constexpr int LDK = K + 2;
    for (int idx = threadIdx.x; idx < 16 * K; idx += 256) {
        const int r = idx / K;
        const int c = idx - r * K;
        lds1[r * LDK + c] = A1[(long long)(m0 + r) * K + c];
        lds2[r * LDK + c] = A2[(long long)(m0 + r) * K + c] * invdeg[m0 + r];
    }
    __syncthreads();
}

template <int K>
__global__ void sage_gemm_store(const float* __restrict__ A1, const float* __restrict__ A2,
                                const float* __restrict__ invdeg,
                                const float* __restrict__ Ws, const float* __restrict__ Wn,
                                const float* __restrict__ bias, float* __restrict__ out) {
    constexpr int LDK = K + 2;
    __shared__ float lds1[16 * LDK];
    __shared__ float lds2[16 * LDK];
    const int m0 = blockIdx.x * 16;
    sage_stage<K>(lds1, lds2, A1, A2, invdeg, m0);

    const int wave = threadIdx.x >> 5;
    const int lane = threadIdx.x & 31;
    const int half = lane >> 4;
    const int l16  = lane & 15;
    const int n0   = wave * 16;
    v8f acc = sage_tile_lds<K>(lds1, lds2, Ws, Wn, n0);
    const int bc = n0 + l16;
    const float bval = bias[bc];
    #pragma unroll
    for (int r = 0; r < 8; ++r) {
        const int row = m0 + r + 8 * half;
        out[(long long)row * GNN_H + bc] = fmaxf(acc[r] + bval, 0.0f);
    }
}

// Layer-2 variant: fuse relu + per-graph sum readout (atomicAdd into gnn_cat),
// eliminating the 51 MB h2 buffer and a full readout pass.
template <int K>
__global__ void sage_gemm_readout(const float* __restrict__ A1, const float* __restrict__ A2,
                                  const float* __restrict__ invdeg,
                                  const float* __restrict__ Ws, const float* __restrict__ Wn,
                                  const float* __restrict__ bias,
                                  const int* __restrict__ gids,
                                  float* __restrict__ gnn_cat) {
    constexpr int LDK = K + 2;
    __shared__ float lds1[16 * LDK];
    __shared__ float lds2[16 * LDK];
    const int m0 = blockIdx.x * 16;
    sage_stage<K>(lds1, lds2, A1, A2, invdeg, m0);

    const int wave = threadIdx.x >> 5;
    const int lane = threadIdx.x & 31;
    const int half = lane >> 4;
    const int l16  = lane & 15;
    const int n0   = wave * 16;
    v8f acc = sage_tile_lds<K>(lds1, lds2, Ws, Wn, n0);
    const int bc = n0 + l16;
    const float bval = bias[bc];
    #pragma unroll
    for (int r = 0; r < 8; ++r) {
        const int row = m0 + r + 8 * half;
        const float v = fmaxf(acc[r] + bval, 0.0f);
        atomicAdd(&gnn_cat[gids[row] * CAT_W + bc], v);
    }
}

// ---------------------------------------------------------------------------
// Dense FC GEMM: C = act(A[M,K] @ W[K,N] + b). One wave per 16x16 tile.
// Operands are tiny (<=1MB, L2-resident); direct loads are fine.
// ---------------------------------------------------------------------------
__global__ void fc_gemm(const float* __restrict__ A, const float* __restrict__ W,
                        const float* __restrict__ bias, float* __restrict__ C,
                        int K, int N, int NT, int do_relu) {
    const int wid  = blockIdx.x * (blockDim.x >> 5) + (threadIdx.x >> 5);
    const int mt   = wid / NT;
    const int nt   = wid - mt * NT;
    const int lane = threadIdx.x & 31;
    const int half = lane >> 4;
    const int l16  = lane & 15;
    const int m0 = mt * 16, n0 = nt * 16;
    const int ar = m0 + l16, bc = n0 + l16;
    v8f acc = {0.f, 0.f, 0.f, 0.f, 0.f, 0.f, 0.f, 0.f};
    const float* a = A + (long long)ar * K;
    for (int k0 = 0; k0 < K; k0 += 4) {
        const int ka = k0 + 2 * half;
        acc = wmma4(a[ka], a[ka + 1], W[ka * N + bc], W[(ka + 1) * N + bc], acc);
    }
    const float bval = bias[bc];
    #pragma unroll
    for (int r = 0; r < 8; ++r) {
        const int row = m0 + r + 8 * half;
        float v = acc[r] + bval;
        if (do_relu) v = fmaxf(v, 0.0f);
        C[(long long)row * N + bc] = v;
    }
}

// Final prediction: out[g] = softplus(-(fc2_out[g,:] . pred_W + pred_b))
__global__ void pred_kernel(const float* __restrict__ A, const float* __restrict__ w,
                            const float* __restrict__ b, float* __restrict__ out) {
    const int g = blockIdx.x * (blockDim.x >> 5) + (threadIdx.x >> 5);
    const int lane = threadIdx.x & 31;
    if (g >= N_GRAPHS) return;
    float s = 0.0f;
    for (int j = lane; j < FC_H; j += 32) s += A[g * FC_H + j] * w[j];
    #pragma unroll
    for (int off = 16; off > 0; off >>= 1) s += __shfl_xor(s, off, 32);
    if (lane == 0) {
        const float v = s + b[0];
        const float t = -v;   // -log_sigmoid(v) = softplus(-v)
        out[g] = fmaxf(t, 0.0f) + log1pf(expf(-fabsf(t)));
    }
}

// ---------------------------------------------------------------------------
// Host-side orchestration
// ---------------------------------------------------------------------------
extern "C" void kernel_launch(void* const* d_in, const int* in_sizes, int n_in,
                              void* d_out, int out_size, void* d_ws, size_t ws_size,
                              hipStream_t stream) {
    const float* x    = (const float*)d_in[0];
    const float* sf   = (const float*)d_in[1];
    const int*   src  = (const int*)d_in[2];
    const int*   dst  = (const int*)d_in[3];
    const int*   gids = (const int*)d_in[4];
    const float* W1s  = (const float*)d_in[5];
    const float* W1n  = (const float*)d_in[6];
    const float* b1   = (const float*)d_in[7];
    const float* W2s  = (const float*)d_in[8];
    const float* W2n  = (const float*)d_in[9];
    const float* b2   = (const float*)d_in[10];
    const float* fc1W = (const float*)d_in[11];
    const float* fc1b = (const float*)d_in[12];
    const float* fc2W = (const float*)d_in[13];
    const float* fc2b = (const float*)d_in[14];
    const float* pW   = (const float*)d_in[15];
    const float* pb   = (const float*)d_in[16];
    float* out = (float*)d_out;

    // Workspace layout (float offsets, 32-float aligned)
    float* ws  = (float*)d_ws;
    float* deg = ws;                                  // 100000 (becomes inv_deg)
    float* agg = ws + 100032;                         // 100000*128 (reused both layers)
    float* h1  = agg + (long long)N_NODES * GNN_H;    // 100000*128
    float* gnn = h1 + (long long)N_NODES * GNN_H;     // 128*132
    float* f1o = gnn + N_GRAPHS * CAT_W + 32;         // 128*512
    float* f2o = f1o + N_GRAPHS * FC_H;               // 128*512

    // Init
    zero_f32<<<1024, 256, 0, stream>>>(deg, N_NODES);
    zero_f32<<<4096, 256, 0, stream>>>(agg, (long long)N_NODES * IN_FEAT);
    zero_f32<<<66, 256, 0, stream>>>(gnn, N_GRAPHS * CAT_W);
    copy_static<<<2, 256, 0, stream>>>(sf, gnn);

    // Degrees -> inverse degrees (in place)
    deg_count<<<(N_EDGES + 255) / 256, 256, 0, stream>>>(dst, deg);
    invdeg_kernel<<<(N_NODES + 255) / 256, 256, 0, stream>>>(deg);

    // Layer 1: scatter + GEMM (K=44) -> h1
    {
        long long tot = (long long)N_EDGES * IN_FEAT;
        scatter_add<IN_FEAT><<<(unsigned)((tot + 255) / 256), 256, 0, stream>>>(x, src, dst, agg);
    }
    sage_gemm_store<IN_FEAT><<<N_NODES / 16, 256, 0, stream>>>(x, agg, deg, W1s, W1n, b1, h1);

    // Layer 2: re-zero agg, scatter h1, GEMM (K=128) fused with graph readout
    zero_f32<<<8192, 256, 0, stream>>>(agg, (long long)N_NODES * GNN_H);
    {
        long long tot = (long long)N_EDGES * GNN_H;
        scatter_add<GNN_H><<<(unsigned)((tot + 255) / 256), 256, 0, stream>>>(h1, src, dst, agg);
    }
    sage_gemm_readout<GNN_H><<<N_NODES / 16, 256, 0, stream>>>(h1, agg, deg, W2s, W2n, b2,
                                                               gids, gnn);

    // MLP head: fc1 (K=132) -> fc2 (K=512) -> pred + softplus
    fc_gemm<<<(N_GRAPHS / 16) * (FC_H / 16) / 8, 256, 0, stream>>>(gnn, fc1W, fc1b, f1o,
                                                                   CAT_W, FC_H, FC_H / 16, 1);
    fc_gemm<<<(N_GRAPHS / 16) * (FC_H / 16) / 8, 256, 0, stream>>>(f1o, fc2W, fc2b, f2o,
                                                                   FC_H, FC_H, FC_H / 16, 1);
    pred_kernel<<<32, 128, 0, stream>>>(f2o, pW, pb, out);
}